// DualMemoryElmanChunkedCell_8151847928496
// MI455X (gfx1250) — compile-verified
//
#include <hip/hip_runtime.h>
#include <math.h>

#define D_DIM   1024
#define B_DIM   8
#define NSLOT   64
#define CHUNKK  64
#define T_DIM   2048
#define NCHUNK  32

typedef float v2f __attribute__((ext_vector_type(2)));
typedef float v8f __attribute__((ext_vector_type(8)));
typedef unsigned int v4u __attribute__((ext_vector_type(4)));
typedef int v8i __attribute__((ext_vector_type(8)));

#define V8F_ZERO {0.0f,0.0f,0.0f,0.0f,0.0f,0.0f,0.0f,0.0f}

__device__ __forceinline__ v8f wmma_f32x4(v2f a, v2f b, v8f c) {
  // D = A(16x4,f32) * B(4x16,f32) + C(16x16,f32)
  return __builtin_amdgcn_wmma_f32_16x16x4_f32(false, a, false, b, (short)0, c,
                                               false, false);
}

// ---------------------------------------------------------------------------
// Kernel 1: Wx[t][b][e] = sum_d x[b,t,d] * W_x[e,d]
// grid 1024 (M-tiles of 16 rows of [B*T]), block 256 (8 waves); each wave does
// 8 N-tiles of 16, K-loop 256 x v_wmma_f32_16x16x4_f32.
// ---------------------------------------------------------------------------
__global__ void wx_gemm_kernel(const float* __restrict__ X,
                               const float* __restrict__ W,
                               float* __restrict__ Wxbuf) {
  const int lane = threadIdx.x & 31;
  const int wave = threadIdx.x >> 5;
  const int lm = lane & 15;
  const int kp = (lane >> 4) << 1;
  const int hi = (lane >> 4) << 3;
  const int m0 = blockIdx.x << 4;

  const float* Ap = X + (size_t)(m0 + lm) * D_DIM + kp;
  for (int t = 0; t < 8; ++t) {
    const int n0 = ((wave << 3) + t) << 4;
    const float* Bp = W + (size_t)(n0 + lm) * D_DIM + kp;
    v8f c = V8F_ZERO;
#pragma unroll 4
    for (int k = 0; k < D_DIM; k += 4)
      c = wmma_f32x4(*(const v2f*)(Ap + k), *(const v2f*)(Bp + k), c);
    const int n = n0 + lm;
#pragma unroll
    for (int r = 0; r < 8; ++r) {
      const int m  = m0 + r + hi;        // m = b*T + t
      const int bb = m >> 11;            // T = 2048
      const int tt = m & (T_DIM - 1);
      Wxbuf[((size_t)tt * B_DIM + bb) * D_DIM + n] = c[r];
    }
  }
}

// ---------------------------------------------------------------------------
// Kernel 2: chunk recurrence. grid 16 WGs (64-dim N-slices), block 128 (4 waves).
// All 8 batches live in one M=16 WMMA tile (hbuf rows 8..15 are zero).
// W_h slice is DMA'd into LDS by the Tensor Data Mover (one tensor_load_to_lds
// per wave, 16 rows each) with pad_enable (1 dword pad / 256 dwords) giving an
// effective row pitch of 1028 dwords -> conflict-free B-fragment reads.
// Per-step inter-WG sync: CDNA5 cluster barrier + agent-scope atomic
// arrive/spin (carries correctness under a plain dispatch where cluster
// barriers are NOP).
// ---------------------------------------------------------------------------
__global__ void recurrence_kernel(const float* __restrict__ Wh,
                                  const float* __restrict__ bh,
                                  const float* __restrict__ Wxbuf,
                                  float* __restrict__ hbuf,      // [16][1024]
                                  float* __restrict__ Hc,        // [64][8][1024]
                                  unsigned* __restrict__ bar,
                                  int chunk) {
  extern __shared__ float wlds[];   // 64 rows x 1028-dword padded pitch (263,168 B)
  const int tid  = threadIdx.x;
  const int lane = tid & 31;
  const int wave = tid >> 5;                      // 0..3
  const int lm = lane & 15;
  const int kp = (lane >> 4) << 1;
  const int n0 = blockIdx.x << 6;                 // this WG's 64-dim slice
  const int wn = wave << 4;                       // wave's 16-dim subtile

  // ---- TDM: DMA W_h[n0+wave*16 .. +15][0..1023] -> LDS (padded) ----
  {
    const unsigned lds_off = (unsigned)(size_t)(void*)wlds +
                             (unsigned)(wave * 16 * 1028 * 4);
    const unsigned long long gaddr =
        (unsigned long long)(size_t)(Wh + (size_t)(n0 + wave * 16) * D_DIM);
    v4u g0;
    g0.x = 1u;                                               // count=1 (user D#)
    g0.y = lds_off;                                          // lds_addr (bytes)
    g0.z = (unsigned)(gaddr & 0xffffffffull);                // global_addr lo
    g0.w = (unsigned)((gaddr >> 32) & 0x1ffffffull) |        // global_addr hi
           (2u << 30);                                       // type=2 (image)
    v8i g1;
    g1[0] = (int)((2u << 16) |      // data_size = 4B
                  (1u << 20) |      // pad_enable
                  (7u << 22) |      // pad_interval: 256 dwords
                  (0u << 25));      // pad_amount: 1 dword
    g1[1] = (int)((1024u & 0xffffu) << 16);                  // tensor_dim0 lo16
    g1[2] = (int)((1024u >> 16) | (16u << 16));              // dim0 hi | dim1 lo
    g1[3] = (int)(1024u << 16);                              // dim1 hi | tile_dim0
    g1[4] = 16;                                              // tile_dim1 | tile_dim2=0
    g1[5] = 1024;                                            // tensor_dim0_stride lo
    g1[6] = 0;                                               // stride hi | dim1_stride
    g1[7] = 0;
    asm volatile("tensor_load_to_lds %0, %1" :: "s"(g0), "s"(g1) : "memory");
    __builtin_amdgcn_s_wait_tensorcnt(0);
  }
  __syncthreads();

  const float* Ap = hbuf + lm * D_DIM + kp;
  const float* wx_base = Wxbuf + (size_t)chunk * CHUNKK * B_DIM * D_DIM;
  const float* wrow = wlds + (wn + lm) * 1028;    // this lane's W_h row in LDS

  for (int step = 0; step < CHUNKK; ++step) {
    // prefetch next step's additive input projection into L2/L0
    __builtin_prefetch(wx_base + (size_t)(step + 1) * B_DIM * D_DIM +
                       (n0 + wn + lane), 0, 1);
    v8f c = V8F_ZERO;
#pragma unroll 4
    for (int k = 0; k < D_DIM; k += 4) {
      v2f a = *(const v2f*)(Ap + k);
      const int q  = k + kp;
      const int qa = q + (q >> 8);                // skip TDM pad dwords
      v2f b;
      b.x = wrow[qa];
      b.y = wrow[qa + 1];
      c = wmma_f32x4(a, b, c);
    }
    if (lane < 16) {                               // rows 0..7 = batches
      const int e = n0 + wn + lane;
      const float bhv = bh[e];
      const float* wx = wx_base + (size_t)step * B_DIM * D_DIM + e;
#pragma unroll
      for (int r = 0; r < 8; ++r) {
        const float v = tanhf(c[r] + wx[(size_t)r * D_DIM] + bhv);
        hbuf[(size_t)r * D_DIM + e] = v;
        Hc[((size_t)step * B_DIM + r) * D_DIM + e] = v;
      }
    }
    // ---- inter-workgroup step barrier (16 WGs) ----
    __threadfence();
    __syncthreads();
    if (tid == 0) {
      __hip_atomic_fetch_add(bar, 1u, __ATOMIC_ACQ_REL, __HIP_MEMORY_SCOPE_AGENT);
      const unsigned target = 16u * ((unsigned)chunk * CHUNKK + (unsigned)step + 1u);
      while (__hip_atomic_load(bar, __ATOMIC_ACQUIRE, __HIP_MEMORY_SCOPE_AGENT) < target)
        __builtin_amdgcn_s_sleep(1);
    }
    __builtin_amdgcn_s_cluster_barrier();          // s_barrier_signal/wait -3
    __syncthreads();
    __threadfence();
  }
}

// ---------------------------------------------------------------------------
// Kernel 3: per-batch attention. grid 8 (batch), block 256 (8 waves).
// scores GEMM (WMMA) -> softmax (shfl_xor) -> read GEMM (WMMA) + out = H+read.
// ---------------------------------------------------------------------------
__global__ void attention_kernel(const float* __restrict__ Hc,
                                 const float* __restrict__ tape,
                                 float* __restrict__ attnbuf,   // [64][8][64]
                                 float* __restrict__ out,
                                 int chunk) {
  __shared__ float sl[64 * 66];
  const int b = blockIdx.x;
  const int tid = threadIdx.x;
  const int lane = tid & 31;
  const int wave = tid >> 5;
  const int lm = lane & 15;
  const int kp = (lane >> 4) << 1;
  const int hi = (lane >> 4) << 3;
  const float scale = 0.03125f;                    // 1/sqrt(1024)

  // Phase A: scores[k,n] = H[k,b,:]·tape[b,n,:] * scale   (16 tiles, 2/wave)
  for (int t = wave * 2; t < wave * 2 + 2; ++t) {
    const int kt = t >> 2, nt = t & 3;
    const float* Ap = Hc   + ((size_t)(kt * 16 + lm) * B_DIM + b) * D_DIM + kp;
    const float* Bp = tape + ((size_t)b * NSLOT + nt * 16 + lm) * D_DIM + kp;
    v8f c = V8F_ZERO;
#pragma unroll 4
    for (int k = 0; k < D_DIM; k += 4)
      c = wmma_f32x4(*(const v2f*)(Ap + k), *(const v2f*)(Bp + k), c);
#pragma unroll
    for (int r = 0; r < 8; ++r)
      sl[(kt * 16 + r + hi) * 66 + nt * 16 + lm] = c[r] * scale;
  }
  __syncthreads();

  // Phase B: softmax over n (64), 8 rows per wave, 2 cols per lane
  for (int j = 0; j < 8; ++j) {
    const int row = wave * 8 + j;
    float x0 = sl[row * 66 + lane];
    float x1 = sl[row * 66 + lane + 32];
    float m = fmaxf(x0, x1);
    for (int off = 16; off > 0; off >>= 1) m = fmaxf(m, __shfl_xor(m, off, 32));
    float e0 = __expf(x0 - m), e1 = __expf(x1 - m);
    float s = e0 + e1;
    for (int off = 16; off > 0; off >>= 1) s += __shfl_xor(s, off, 32);
    const float inv = 1.0f / s;
    e0 *= inv; e1 *= inv;
    sl[row * 66 + lane] = e0;
    sl[row * 66 + lane + 32] = e1;
    attnbuf[((size_t)row * B_DIM + b) * NSLOT + lane] = e0;
    attnbuf[((size_t)row * B_DIM + b) * NSLOT + lane + 32] = e1;
  }
  __syncthreads();

  // Phase C: read[k,d] = attn[k,:]·tape[b,:,d]; out = H + read  (256 tiles)
  for (int i = 0; i < 32; ++i) {
    const int tt = wave + (i << 3);
    const int kt = tt >> 6, dt = tt & 63;
    const int dcol = (dt << 4) + lm;
    v8f c = V8F_ZERO;
#pragma unroll
    for (int kk = 0; kk < NSLOT; kk += 4) {
      v2f a = *(const v2f*)(&sl[(kt * 16 + lm) * 66 + kk + kp]);
      v2f bb;
      bb.x = tape[((size_t)b * NSLOT + kk + kp)     * D_DIM + dcol];
      bb.y = tape[((size_t)b * NSLOT + kk + kp + 1) * D_DIM + dcol];
      c = wmma_f32x4(a, bb, c);
    }
#pragma unroll
    for (int r = 0; r < 8; ++r) {
      const int krow = kt * 16 + r + hi;
      const float v = c[r] + Hc[((size_t)krow * B_DIM + b) * D_DIM + dcol];
      out[((size_t)b * T_DIM + (size_t)chunk * CHUNKK + krow) * D_DIM + dcol] = v;
    }
  }
}

// ---------------------------------------------------------------------------
// Kernel 4: decayed tape write. grid 128 = (batch b, 64-wide e-slice),
// block 256 (8 waves). wv GEMM (WMMA,K=1024) -> reverse-cumprod coeffs ->
// tmp[n,e] GEMM (WMMA,K=64) -> tape = tape*tape0c + tmp (in place, exclusive).
// ---------------------------------------------------------------------------
__global__ void tape_update_kernel(const float* __restrict__ Hc,
                                   const float* __restrict__ Ww,
                                   const float* __restrict__ attnbuf,
                                   float* __restrict__ tape) {
  __shared__ float wv[64 * 66];
  __shared__ float vc[64 * 66];
  __shared__ float t0c[64];
  const int b  = blockIdx.x >> 4;
  const int es = (blockIdx.x & 15) << 6;
  const int tid = threadIdx.x;
  const int lane = tid & 31;
  const int wave = tid >> 5;
  const int lm = lane & 15;
  const int kp = (lane >> 4) << 1;
  const int hi = (lane >> 4) << 3;

  // Phase A: wv[k][el] = H[k,b,:]·W_write[es+el,:]
  for (int t = wave * 2; t < wave * 2 + 2; ++t) {
    const int kt = t >> 2, et = t & 3;
    const float* Ap = Hc + ((size_t)(kt * 16 + lm) * B_DIM + b) * D_DIM + kp;
    const float* Bp = Ww + (size_t)(es + et * 16 + lm) * D_DIM + kp;
    v8f c = V8F_ZERO;
#pragma unroll 4
    for (int k = 0; k < D_DIM; k += 4)
      c = wmma_f32x4(*(const v2f*)(Ap + k), *(const v2f*)(Bp + k), c);
#pragma unroll
    for (int r = 0; r < 8; ++r)
      wv[(kt * 16 + r + hi) * 66 + et * 16 + lm] = c[r];
  }

  // reverse cumprod coefficients per slot n
  if (tid < NSLOT) {
    const int n = tid;
    float prod = 1.0f;
    for (int k = CHUNKK - 1; k >= 0; --k) {
      const float a = attnbuf[((size_t)k * B_DIM + b) * NSLOT + n];
      vc[k * 66 + n] = a * prod;
      prod *= (1.0f - a);
    }
    t0c[n] = prod;
  }
  __syncthreads();

  // Phase B: tmp[n][e] = sum_k vc[k][n]*wv[k][e]; tape update
  for (int t = wave * 2; t < wave * 2 + 2; ++t) {
    const int nt = t >> 2, et = t & 3;
    v8f c = V8F_ZERO;
#pragma unroll
    for (int kk = 0; kk < CHUNKK; kk += 4) {
      v2f a, bb;
      a.x  = vc[(kk + kp)     * 66 + nt * 16 + lm];
      a.y  = vc[(kk + kp + 1) * 66 + nt * 16 + lm];
      bb.x = wv[(kk + kp)     * 66 + et * 16 + lm];
      bb.y = wv[(kk + kp + 1) * 66 + et * 16 + lm];
      c = wmma_f32x4(a, bb, c);
    }
#pragma unroll
    for (int r = 0; r < 8; ++r) {
      const int n = nt * 16 + r + hi;
      const int e = es + et * 16 + lm;
      const size_t idx = ((size_t)b * NSLOT + n) * D_DIM + e;
      tape[idx] = tape[idx] * t0c[n] + c[r];
    }
  }
}

// ---------------------------------------------------------------------------
extern "C" void kernel_launch(void* const* d_in, const int* in_sizes, int n_in,
                              void* d_out, int out_size, void* d_ws, size_t ws_size,
                              hipStream_t stream) {
  (void)in_sizes; (void)n_in; (void)out_size; (void)ws_size;
  const float* x_seq  = (const float*)d_in[0];
  const float* h_tape = (const float*)d_in[1];
  const float* h_work = (const float*)d_in[2];
  const float* W_h    = (const float*)d_in[3];
  const float* W_x    = (const float*)d_in[4];
  const float* b_h    = (const float*)d_in[5];
  const float* W_wr   = (const float*)d_in[6];

  float* out  = (float*)d_out;                                   // [B,T,D]
  float* tape = out + (size_t)B_DIM * T_DIM * D_DIM;             // [B,N,D]
  float* hwo  = tape + (size_t)B_DIM * NSLOT * D_DIM;            // [B,D]

  float* w      = (float*)d_ws;
  float* Wxbuf  = w;                                             // 16,777,216 f
  float* Hc     = Wxbuf + (size_t)16777216;                      //    524,288 f
  float* attnb  = Hc + 524288;                                   //     32,768 f
  float* hbuf   = attnb + 32768;                                 //     16,384 f
  unsigned* bar = (unsigned*)(hbuf + 16384);

  hipMemsetAsync(hbuf, 0, 16384 * sizeof(float), stream);        // rows 8..15 = 0
  hipMemsetAsync(bar, 0, 64, stream);
  hipMemcpyAsync(hbuf, h_work, (size_t)B_DIM * D_DIM * sizeof(float),
                 hipMemcpyDeviceToDevice, stream);
  hipMemcpyAsync(tape, h_tape, (size_t)B_DIM * NSLOT * D_DIM * sizeof(float),
                 hipMemcpyDeviceToDevice, stream);

  wx_gemm_kernel<<<dim3(1024), dim3(256), 0, stream>>>(x_seq, W_x, Wxbuf);

  const size_t wlds_bytes = 64 * 1028 * sizeof(float);           // 263,168 B
  for (int c = 0; c < NCHUNK; ++c) {
    recurrence_kernel<<<dim3(16), dim3(128), wlds_bytes, stream>>>(
        W_h, b_h, Wxbuf, hbuf, Hc, bar, c);
    attention_kernel<<<dim3(8), dim3(256), 0, stream>>>(Hc, tape, attnb, out, c);
    tape_update_kernel<<<dim3(128), dim3(256), 0, stream>>>(Hc, W_wr, attnb, tape);
  }

  hipMemcpyAsync(hwo, hbuf, (size_t)B_DIM * D_DIM * sizeof(float),
                 hipMemcpyDeviceToDevice, stream);
}